// BilinearResampling_34419867910153
// MI455X (gfx1250) — compile-verified
//
#include <hip/hip_runtime.h>
#include <hip/hip_bf16.h>

// Bilinear warp (grid-sample with flow field), clamp-to-edge.
// x: [B=8, C=8, H=1024, W=1024] f32;  warp: [B, 2, H, W] f32 (dx plane, dy plane)
// out[b,c,y,w] = bilerp(x[b,c], fx = clamp(w + dx, 0, W-1), fy = clamp(y + dy, 0, H-1))
//
// Memory-bound gather (~576 MB @ 23.3 TB/s ~= 25 us). No matmul structure -> no WMMA.
// v2: two adjacent pixels per thread ->
//   - warp read: one aligned b64 NT load per plane (dx pair, dy pair)
//   - output:    one aligned b64 NT store per channel (2 outputs/store)
//   - coordinate/weight setup amortized over 16 outputs
//   - corner gathers keep constant deltas (+1, +W, +W+1) so the compiler merges
//     them into b64 loads with immediate offsets (verified in round-1 disasm)
// Explicit prefetches removed: the fully-unrolled loop already clauses all 32
// gather loads in flight (LOADcnt depth 63), so prefetch was redundant issue.

namespace {
constexpr int kB   = 8;
constexpr int kC   = 8;
constexpr int kH   = 1024;
constexpr int kW   = 1024;
constexpr int kHW  = kH * kW;        // 1,048,576 elements (4 MB plane)
constexpr int kCHW = kC * kHW;       // per-batch x/out stride
}

typedef float v2f __attribute__((ext_vector_type(2)));

__global__ __launch_bounds__(256) void
BilinearResampling_34419867910153_kernel(const float* __restrict__ x,
                                         const float* __restrict__ warp,
                                         float* __restrict__ out) {
    const int tid = blockIdx.x * 256 + threadIdx.x;   // 0 .. B*H*W/2 - 1
    const int w2 = tid & (kW / 2 - 1);                // pixel-pair index in row
    const int y  = (tid >> 9) & (kH - 1);
    const int b  = tid >> 19;
    const int w0 = w2 << 1;                           // even pixel column

    // ---- flow field: one aligned b64 NT load per plane ----
    const float* wp = warp + (size_t)b * (2 * kHW) + ((size_t)y << 10) + w0;
    const v2f dx = __builtin_nontemporal_load((const v2f*)wp);
    const v2f dy = __builtin_nontemporal_load((const v2f*)(wp + kHW));

    // ---- continuous coords, clamp-to-edge (two pixels: a = w0, b = w0+1) ----
    const float yf = (float)y;
    float fxa = fminf(fmaxf(dx.x + (float)w0,       0.0f), (float)(kW - 1));
    float fxb = fminf(fmaxf(dx.y + (float)(w0 + 1), 0.0f), (float)(kW - 1));
    float fya = fminf(fmaxf(dy.x + yf,              0.0f), (float)(kH - 1));
    float fyb = fminf(fmaxf(dy.y + yf,              0.0f), (float)(kH - 1));

    // Clamp lower corner to W-2/H-2 so x1=x0+1, y1=y0+1 always (identical to the
    // reference at the edge: wx -> 1 selects the last column/row).
    const int x0a = min((int)floorf(fxa), kW - 2);
    const int y0a = min((int)floorf(fya), kH - 2);
    const int x0b = min((int)floorf(fxb), kW - 2);
    const int y0b = min((int)floorf(fyb), kH - 2);
    const float wxa = fxa - (float)x0a, wya = fya - (float)y0a;
    const float wxb = fxb - (float)x0b, wyb = fyb - (float)y0b;

    // 4 bilinear weights per pixel in 5 VALU ops
    const float w11a = wxa * wya;
    const float w10a = wya - w11a;           // (1-wx)*wy
    const float w01a = wxa - w11a;           // wx*(1-wy)
    const float w00a = 1.0f - wxa - w10a;    // (1-wx)*(1-wy)
    const float w11b = wxb * wyb;
    const float w10b = wyb - w11b;
    const float w01b = wxb - w11b;
    const float w00b = 1.0f - wxb - w10b;

    // Gather bases: corners at +0, +1, +W, +W+1 (constant immediate offsets)
    const float* __restrict__ pa = x + (size_t)b * kCHW + ((size_t)y0a << 10) + x0a;
    const float* __restrict__ pb = x + (size_t)b * kCHW + ((size_t)y0b << 10) + x0b;
    float*       __restrict__ q  = out + (size_t)b * kCHW + ((size_t)y << 10) + w0;

#pragma unroll
    for (int c = 0; c < kC; ++c) {
        const float ra = fmaf(pa[0], w00a,
                         fmaf(pa[1], w01a,
                         fmaf(pa[kW], w10a, pa[kW + 1] * w11a)));
        const float rb = fmaf(pb[0], w00b,
                         fmaf(pb[1], w01b,
                         fmaf(pb[kW], w10b, pb[kW + 1] * w11b)));
        const v2f r = {ra, rb};
        // out is write-once: non-temporal b64 store keeps L2 free for x gathers
        __builtin_nontemporal_store(r, (v2f*)q);
        pa += kHW;
        pb += kHW;
        q  += kHW;
    }
}

extern "C" void kernel_launch(void* const* d_in, const int* in_sizes, int n_in,
                              void* d_out, int out_size, void* d_ws, size_t ws_size,
                              hipStream_t stream) {
    const float* x    = (const float*)d_in[0];   // [8,8,1024,1024] f32
    const float* warp = (const float*)d_in[1];   // [8,2,1024,1024] f32
    float*       out  = (float*)d_out;           // [8,8,1024,1024] f32
    (void)in_sizes; (void)n_in; (void)out_size; (void)d_ws; (void)ws_size;

    const int threads = kB * kH * kW / 2;        // one thread per pixel pair
    BilinearResampling_34419867910153_kernel<<<threads / 256, 256, 0, stream>>>(x, warp, out);
}